// SAGEResidualReranker_48885317763305
// MI455X (gfx1250) — compile-verified
//
#include <hip/hip_runtime.h>
#include <math.h>

// CDNA5 / gfx1250. wave32. Uses V_WMMA_F32_16X16X4_F32 for the fp32 GEMMs.

typedef __attribute__((ext_vector_type(2))) float v2f;
typedef __attribute__((ext_vector_type(8))) float v8f;

#define N_NODES 100000
#define N_EDGES 500000
#define IN_DIM  256
#define HID_DIM 128
#define TILE_M  16
#define LDS_STRIDE 260  // 260 % 64 == 4 dwords -> conflict-free half/K read pattern

// ---------------------------------------------------------------------------
// Kernel 0: zero agg_sum + deg in workspace
// ---------------------------------------------------------------------------
__global__ void zero_ws_kernel(float* __restrict__ ws, long long n) {
  long long i = (long long)blockIdx.x * blockDim.x + threadIdx.x;
  long long stride = (long long)gridDim.x * blockDim.x;
  for (; i < n; i += stride) ws[i] = 0.0f;
}

// ---------------------------------------------------------------------------
// Kernel 1: edge scatter-add (mean aggregation numerator + degree)
// 64 threads per edge, each handles 4 consecutive features (float4 gather).
// Atomics resolve in L2 (GLOBAL_ATOMIC_ADD_F32); x and agg_sum both fit in
// the 192MB L2, so this runs at L2 rate, not HBM rate.
// ---------------------------------------------------------------------------
__global__ void __launch_bounds__(256)
edge_scatter_kernel(const float* __restrict__ x,
                    const int*  __restrict__ src,
                    const int*  __restrict__ dst,
                    float* __restrict__ aggsum,
                    float* __restrict__ deg) {
  long long t = (long long)blockIdx.x * blockDim.x + threadIdx.x;
  int e = (int)(t >> 6);
  if (e >= N_EDGES) return;
  int f = ((int)t & 63) << 2;
  int s = src[e];
  int d = dst[e];
  const float4 v = *(const float4*)(x + (long long)s * IN_DIM + f);
  float* o = aggsum + (long long)d * IN_DIM + f;
  atomicAdd(o + 0, v.x);
  atomicAdd(o + 1, v.y);
  atomicAdd(o + 2, v.z);
  atomicAdd(o + 3, v.w);
  if ((t & 63) == 0) atomicAdd(deg + d, 1.0f);
}

// ---------------------------------------------------------------------------
// Kernel 2: fused  mean -> SAGEConv (2 GEMMs) -> ReLU -> residual GEMM ->
//           score head -> sigmoid blend.  h never touches memory.
//
// Block: 256 threads = 8 wave32. Block computes 16 nodes x 128 hidden.
// Wave w owns hidden columns [16w, 16w+16). Per K=4 step each wave issues
// three v_wmma_f32_16x16x4_f32.
//
// fp32 WMMA fragment layout (ISA 7.12.2, 32-bit A 16x4):
//   lane<16: M=lane,    VGPR0 -> K+0, VGPR1 -> K+1
//   lane>=16:M=lane-16, VGPR0 -> K+2, VGPR1 -> K+3
// B (4x16) mirrors with N=lane%16.
// C/D: VGPR v, lane l: row M = v + 8*(l>=16), col N = l%16.
// ---------------------------------------------------------------------------
__global__ void __launch_bounds__(256)
sage_fused_kernel(const float* __restrict__ x,
                  const float* __restrict__ aggsum,
                  const float* __restrict__ deg,
                  const float* __restrict__ Wl,  const float* __restrict__ bl,
                  const float* __restrict__ Wr,
                  const float* __restrict__ Wrp, const float* __restrict__ brp,
                  const float* __restrict__ shw, const float* __restrict__ shb,
                  const float* __restrict__ alpha,
                  const float* __restrict__ rr,
                  float* __restrict__ out) {
  __shared__ float xs[TILE_M][LDS_STRIDE];
  __shared__ float as[TILE_M][LDS_STRIDE];
  __shared__ float scorepart[8][TILE_M];

  const int rowbase = blockIdx.x * TILE_M;
  const int tid = threadIdx.x;

  // ---- stage x tile and mean-aggregated tile into LDS (coalesced float4) ----
  #pragma unroll
  for (int i = 0; i < 4; ++i) {
    int idx = tid + 256 * i;        // 1024 float4 slots per tile
    int r   = idx >> 6;             // 64 float4 per row
    int c4  = (idx & 63) << 2;
    long long g = (long long)(rowbase + r) * IN_DIM + c4;
    float4 xv = *(const float4*)(x + g);
    *(float4*)&xs[r][c4] = xv;
    float4 av = *(const float4*)(aggsum + g);
    float invd = 1.0f / fmaxf(deg[rowbase + r], 1.0f);
    av.x *= invd; av.y *= invd; av.z *= invd; av.w *= invd;
    *(float4*)&as[r][c4] = av;
  }
  __syncthreads();

  const int wave = tid >> 5;
  const int lane = tid & 31;
  const int hf   = lane >> 4;       // lane half (0/1)
  const int m    = lane & 15;
  const int jcol = wave * 16 + m;   // hidden column this lane touches in B/C/D

  const float* wlrow = Wl  + (long long)jcol * IN_DIM;
  const float* wrrow = Wr  + (long long)jcol * IN_DIM;
  const float* wprow = Wrp + (long long)jcol * IN_DIM;

  v8f acc1 = {};  // agg @ Wl^T + x @ Wr^T
  v8f acc2 = {};  // x @ Wrp^T

  #pragma unroll 4
  for (int k = 0; k < IN_DIM; k += 4) {
    const int ko = k + 2 * hf;
    float2 ax = *(const float2*)&xs[m][ko];
    float2 aa = *(const float2*)&as[m][ko];
    float2 wl = *(const float2*)(wlrow + ko);
    float2 wr = *(const float2*)(wrrow + ko);
    float2 wp = *(const float2*)(wprow + ko);
    v2f Aagg = {aa.x, aa.y};
    v2f Ax   = {ax.x, ax.y};
    v2f Bl   = {wl.x, wl.y};
    v2f Br   = {wr.x, wr.y};
    v2f Bp   = {wp.x, wp.y};
    acc1 = __builtin_amdgcn_wmma_f32_16x16x4_f32(false, Aagg, false, Bl,
                                                 (short)0, acc1, false, false);
    acc1 = __builtin_amdgcn_wmma_f32_16x16x4_f32(false, Ax,   false, Br,
                                                 (short)0, acc1, false, false);
    acc2 = __builtin_amdgcn_wmma_f32_16x16x4_f32(false, Ax,   false, Bp,
                                                 (short)0, acc2, false, false);
  }

  // ---- epilogue: bias + ReLU + residual + score-head dot, all in regs ----
  const float blv  = bl[jcol];
  const float brpv = brp[jcol];
  const float swv  = shw[jcol];

  float part[8];
  #pragma unroll
  for (int v = 0; v < 8; ++v) {
    float h = fmaxf(acc1[v] + blv, 0.0f) + acc2[v] + brpv;
    part[v] = h * swv;           // contribution of column jcol to row score
  }
  // reduce the 16 columns of each row (lanes within a 16-lane half)
  #pragma unroll
  for (int v = 0; v < 8; ++v) {
    #pragma unroll
    for (int off = 1; off < 16; off <<= 1)
      part[v] += __shfl_xor(part[v], off, 32);
  }
  // lane 0 holds rows 0..7 (v), lane 16 holds rows 8..15
  if (m == 0) {
    #pragma unroll
    for (int v = 0; v < 8; ++v)
      scorepart[wave][hf * 8 + v] = part[v];
  }
  __syncthreads();

  if (tid < TILE_M) {
    float s = shb[0];
    #pragma unroll
    for (int w = 0; w < 8; ++w) s += scorepart[w][tid];
    float a = 1.0f / (1.0f + __expf(-alpha[0]));
    int node = rowbase + tid;
    out[node] = a * rr[node] + (1.0f - a) * s;
  }
}

// ---------------------------------------------------------------------------
extern "C" void kernel_launch(void* const* d_in, const int* in_sizes, int n_in,
                              void* d_out, int out_size, void* d_ws, size_t ws_size,
                              hipStream_t stream) {
  const float* x     = (const float*)d_in[0];
  const int*   ei    = (const int*)d_in[1];   // (2, N_EDGES): row0=src, row1=dst
  const float* rr    = (const float*)d_in[2];
  const float* Wl    = (const float*)d_in[3];
  const float* bl    = (const float*)d_in[4];
  const float* Wr    = (const float*)d_in[5];
  const float* Wrp   = (const float*)d_in[6];
  const float* brp   = (const float*)d_in[7];
  const float* shw   = (const float*)d_in[8];
  const float* shb   = (const float*)d_in[9];
  const float* alpha = (const float*)d_in[10];

  float* aggsum = (float*)d_ws;                       // N_NODES * IN_DIM
  float* deg    = aggsum + (long long)N_NODES * IN_DIM; // N_NODES

  long long nzero = (long long)N_NODES * IN_DIM + N_NODES;
  zero_ws_kernel<<<2048, 256, 0, stream>>>(aggsum, nzero);

  const int* src = ei;
  const int* dst = ei + N_EDGES;
  long long nthreads = (long long)N_EDGES * 64;
  int sblocks = (int)((nthreads + 255) / 256);
  edge_scatter_kernel<<<sblocks, 256, 0, stream>>>(x, src, dst, aggsum, deg);

  sage_fused_kernel<<<N_NODES / TILE_M, 256, 0, stream>>>(
      x, aggsum, deg, Wl, bl, Wr, Wrp, brp, shw, shb, alpha, rr, (float*)d_out);
}